// GNNWithAdversarialAugmentation_87222195847284
// MI455X (gfx1250) — compile-verified
//
#include <hip/hip_runtime.h>
#include <math.h>

#define NFEAT 128

typedef float v2f __attribute__((ext_vector_type(2)));
typedef float v8f __attribute__((ext_vector_type(8)));
typedef int vec4i __attribute__((vector_size(16)));
typedef __attribute__((address_space(1))) vec4i gvec4i;  // global int32x4
typedef __attribute__((address_space(3))) vec4i svec4i;  // LDS int32x4

// ---------------------------------------------------------------- utilities

__global__ void zero_int_kernel(int* __restrict__ p, int n) {
  int i = blockIdx.x * blockDim.x + threadIdx.x;
  if (i < n) p[i] = 0;
}

// Histogram of destination degrees (int atomics, one-time cost).
__global__ void deg_kernel(const long long* __restrict__ dst,
                           int* __restrict__ deg, int E) {
  int e = blockIdx.x * blockDim.x + threadIdx.x;
  if (e < E) atomicAdd(&deg[(int)dst[e]], 1);
}

// Single-workgroup exclusive scan over N degrees -> rowstart/cursor, plus
// inv[n] = 1/max(deg,1). N=10000 -> 10 chunks of 1024, trivial.
__global__ __launch_bounds__(1024) void scan_kernel(
    const int* __restrict__ deg, int* __restrict__ rowstart,
    int* __restrict__ cursor, float* __restrict__ inv, int N) {
  __shared__ int sh[1024];
  __shared__ int sbase;
  if (threadIdx.x == 0) sbase = 0;
  __syncthreads();
  for (int base = 0; base < N; base += 1024) {
    const int i = base + threadIdx.x;
    const int d = (i < N) ? deg[i] : 0;
    sh[threadIdx.x] = d;
    __syncthreads();
    for (int off = 1; off < 1024; off <<= 1) {
      const int v = sh[threadIdx.x];
      const int add = (threadIdx.x >= off) ? sh[threadIdx.x - off] : 0;
      __syncthreads();
      sh[threadIdx.x] = v + add;
      __syncthreads();
    }
    if (i < N) {
      const int excl = sh[threadIdx.x] - d + sbase;
      rowstart[i] = excl;
      cursor[i] = excl;
      inv[i] = 1.0f / fmaxf((float)d, 1.0f);
    }
    __syncthreads();
    if (threadIdx.x == 0) sbase += sh[1023];
    __syncthreads();
  }
  if (threadIdx.x == 0) rowstart[N] = sbase;
}

// Bucket source indices by destination (CSR fill).
__global__ void fill_kernel(const long long* __restrict__ src,
                            const long long* __restrict__ dst,
                            int* __restrict__ cursor, int* __restrict__ eidx,
                            int E) {
  int e = blockIdx.x * blockDim.x + threadIdx.x;
  if (e < E) {
    const int pos = atomicAdd(&cursor[(int)dst[e]], 1);
    eidx[pos] = (int)src[e];
  }
}

// ------------------------------------------------------- neighbor aggregation
// Atomic-free: one wave per destination node, lane l owns channels [4l,4l+4).
// Streams neighbor rows (512B coalesced per wave, L2-resident), accumulates
// in registers, writes the normalized mean once.
__global__ __launch_bounds__(256) void gather_mean_kernel(
    const float* __restrict__ h, const int* __restrict__ rowstart,
    const int* __restrict__ eidx, const float* __restrict__ inv,
    float* __restrict__ mean, int N) {
  const int n = blockIdx.x * 8 + (threadIdx.x >> 5);
  if (n >= N) return;
  const int lane = threadIdx.x & 31;
  int j0 = __builtin_amdgcn_readfirstlane(rowstart[n]);
  int j1 = __builtin_amdgcn_readfirstlane(rowstart[n + 1]);
  float4 acc = {0.0f, 0.0f, 0.0f, 0.0f};
  for (int j = j0; j < j1; ++j) {
    if (j + 2 < j1)  // hide L2 gather latency
      __builtin_prefetch(h + (size_t)eidx[j + 2] * NFEAT, 0, 1);
    const int s = eidx[j];
    const float4 v = *(const float4*)(h + (size_t)s * NFEAT + lane * 4);
    acc.x += v.x;
    acc.y += v.y;
    acc.z += v.z;
    acc.w += v.w;
  }
  const float sc = inv[n];
  float4 m = {acc.x * sc, acc.y * sc, acc.z * sc, acc.w * sc};
  *(float4*)(mean + (size_t)n * NFEAT + lane * 4) = m;
}

// --------------------------------------------------------------- SAGE GEMM
// out[r, :] = relu( mean[r,:] @ Wl + Xin[r,:] @ Wr + bl )
// One wave per 16-row tile, 8 waves/block. Wl/Wr staged in LDS (128 KB of the
// 320 KB WGP pool) via gfx1250 async LDS copies when available.
__global__ __launch_bounds__(256) void sage_gemm_kernel(
    const float* __restrict__ mean, const float* __restrict__ Xin,
    const float* __restrict__ Wl, const float* __restrict__ bl,
    const float* __restrict__ Wr, float* __restrict__ out, int N) {
  __shared__ float sWl[NFEAT * NFEAT];  // 64 KB
  __shared__ float sWr[NFEAT * NFEAT];  // 64 KB

  const int tid = threadIdx.x;
#if defined(__gfx1250__) && \
    __has_builtin(__builtin_amdgcn_global_load_async_to_lds_b128)
  // ASYNCcnt-tracked DMA: 16B per lane per op, no VGPR round-trip.
  for (int i = tid * 4; i < NFEAT * NFEAT; i += 256 * 4) {
    __builtin_amdgcn_global_load_async_to_lds_b128(
        (gvec4i*)(Wl + i), (svec4i*)(sWl + i), 0, 0);
    __builtin_amdgcn_global_load_async_to_lds_b128(
        (gvec4i*)(Wr + i), (svec4i*)(sWr + i), 0, 0);
  }
#if __has_builtin(__builtin_amdgcn_s_wait_asynccnt)
  __builtin_amdgcn_s_wait_asynccnt(0);
#else
  asm volatile("s_wait_asynccnt 0" ::: "memory");
#endif
#else
  for (int i = tid * 4; i < NFEAT * NFEAT; i += 256 * 4) {
    *(float4*)(sWl + i) = *(const float4*)(Wl + i);
    *(float4*)(sWr + i) = *(const float4*)(Wr + i);
  }
#endif
  __syncthreads();

  const int wave = tid >> 5;
  const int lane = tid & 31;
  const int rt = blockIdx.x * 8 + wave;  // 16-row tile (wave-uniform)
  if (rt * 16 >= N) return;              // whole wave exits together

  const int lo = lane & 15;  // M (A) / N (B,C,D)
  const int hi = lane >> 4;  // selects K pair within a K=4 step
  const int row = rt * 16 + lo;

  // A fragments for both operands: V0 = {K0|K2}, V1 = {K1|K3} across halves.
  v2f aM[32], aX[32];
  const float* mRow = mean + (size_t)row * NFEAT;
  const float* xRow = Xin + (size_t)row * NFEAT;
#pragma unroll
  for (int k = 0; k < 32; ++k) {
    const int kb = k * 4 + hi * 2;
    aM[k].x = mRow[kb];
    aM[k].y = mRow[kb + 1];
    aX[k].x = xRow[kb];
    aX[k].y = xRow[kb + 1];
  }

  for (int nt = 0; nt < 8; ++nt) {
    const int col = nt * 16 + lo;
    const float bv = bl[col];
    v8f acc = {bv, bv, bv, bv, bv, bv, bv, bv};
#pragma unroll
    for (int k = 0; k < 32; ++k) {
      const int kb = k * 4 + hi * 2;
      v2f b;
      b.x = sWl[kb * NFEAT + col];
      b.y = sWl[(kb + 1) * NFEAT + col];
      acc = __builtin_amdgcn_wmma_f32_16x16x4_f32(
          false, aM[k], false, b, (short)0, acc, false, false);
    }
#pragma unroll
    for (int k = 0; k < 32; ++k) {
      const int kb = k * 4 + hi * 2;
      v2f b;
      b.x = sWr[kb * NFEAT + col];
      b.y = sWr[(kb + 1) * NFEAT + col];
      acc = __builtin_amdgcn_wmma_f32_16x16x4_f32(
          false, aX[k], false, b, (short)0, acc, false, false);
    }
    // C/D layout: vgpr v holds M = v + 8*hi, N = lo.
#pragma unroll
    for (int v = 0; v < 8; ++v) {
      const int r = rt * 16 + hi * 8 + v;
      out[(size_t)r * NFEAT + col] = fmaxf(acc[v], 0.0f);
    }
  }
}

// ------------------------------------------------------------- classifier
__global__ void cls_kernel(const float* __restrict__ h,
                           const float* __restrict__ Wc,
                           const float* __restrict__ bc,
                           float* __restrict__ out, int N) {
  int i = blockIdx.x * blockDim.x + threadIdx.x;
  if (i >= N) return;
  const float* hr = h + (size_t)i * NFEAT;
  float s = bc[0];
#pragma unroll 8
  for (int k = 0; k < NFEAT; ++k) s += hr[k] * Wc[k];
  out[i] = 1.0f / (1.0f + expf(-s));
}

// ---------------------------------------------------------------- launcher

extern "C" void kernel_launch(void* const* d_in, const int* in_sizes, int n_in,
                              void* d_out, int out_size, void* d_ws,
                              size_t ws_size, hipStream_t stream) {
  const float* x = (const float*)d_in[0];
  const long long* ei = (const long long*)d_in[1];  // int64 [2, E]
  const float* W1l = (const float*)d_in[2];
  const float* b1l = (const float*)d_in[3];
  const float* W1r = (const float*)d_in[4];
  const float* W2l = (const float*)d_in[5];
  const float* b2l = (const float*)d_in[6];
  const float* W2r = (const float*)d_in[7];
  const float* W3l = (const float*)d_in[8];
  const float* b3l = (const float*)d_in[9];
  const float* W3r = (const float*)d_in[10];
  const float* Wc = (const float*)d_in[11];
  const float* bc = (const float*)d_in[12];
  float* out = (float*)d_out;

  const int N = in_sizes[0] / NFEAT;  // 10000
  const int E = in_sizes[1] / 2;      // 640000
  const long long* src = ei;
  const long long* dst = ei + E;

  // Workspace carve-up (256B-aligned bump allocator).
  char* p = (char*)d_ws;
  auto alloc = [&](size_t bytes) -> char* {
    char* r = p;
    p += (bytes + 255) & ~(size_t)255;
    return r;
  };
  const size_t NF = (size_t)N * NFEAT;
  float* inv = (float*)alloc((size_t)N * 4);
  int* deg = (int*)alloc((size_t)N * 4);
  int* rowstart = (int*)alloc((size_t)(N + 1) * 4);
  int* cursor = (int*)alloc((size_t)N * 4);
  int* eidx = (int*)alloc((size_t)E * 4);
  float* mean = (float*)alloc(NF * 4);
  float* h1 = (float*)alloc(NF * 4);
  float* h2 = (float*)alloc(NF * 4);
  (void)ws_size;

  const int tpb = 256;
  const int nBlk = (N + tpb - 1) / tpb;
  const int eBlk = (E + tpb - 1) / tpb;
  const int wBlk = (N + 7) / 8;   // 1 wave per node, 8 waves/block
  const int NT = (N + 15) / 16;   // 16-row tiles
  const int gBlk = (NT + 7) / 8;  // 8 tiles (waves) per block

  // --- One-time CSR build (reused by all 3 layers) ---
  zero_int_kernel<<<nBlk, tpb, 0, stream>>>(deg, N);
  deg_kernel<<<eBlk, tpb, 0, stream>>>(dst, deg, E);
  scan_kernel<<<1, 1024, 0, stream>>>(deg, rowstart, cursor, inv, N);
  fill_kernel<<<eBlk, tpb, 0, stream>>>(src, dst, cursor, eidx, E);

  // Layer 1: x -> h1
  gather_mean_kernel<<<wBlk, tpb, 0, stream>>>(x, rowstart, eidx, inv, mean, N);
  sage_gemm_kernel<<<gBlk, tpb, 0, stream>>>(mean, x, W1l, b1l, W1r, h1, N);

  // Layer 2: h1 -> h2
  gather_mean_kernel<<<wBlk, tpb, 0, stream>>>(h1, rowstart, eidx, inv, mean, N);
  sage_gemm_kernel<<<gBlk, tpb, 0, stream>>>(mean, h1, W2l, b2l, W2r, h2, N);

  // Layer 3: h2 -> h1
  gather_mean_kernel<<<wBlk, tpb, 0, stream>>>(h2, rowstart, eidx, inv, mean, N);
  sage_gemm_kernel<<<gBlk, tpb, 0, stream>>>(mean, h2, W3l, b3l, W3r, h1, N);

  // Classifier + sigmoid.
  cls_kernel<<<nBlk, tpb, 0, stream>>>(h1, Wc, bc, out, N);
}